// DiceCEIgnoreLoss_6511170421565
// MI455X (gfx1250) — compile-verified
//
#include <hip/hip_runtime.h>

// DiceCE loss, MI455X (gfx1250, wave32).
// Pass 1: streaming partial reduction (memory-bound, ~160 MiB @ 23.3 TB/s ≈ 7 us floor).
// Pass 2: deterministic finalize. Wave reductions use V_WMMA_F32_16X16X4_F32 (ones-matmul allreduce).

typedef float v2f __attribute__((ext_vector_type(2)));
typedef float v8f __attribute__((ext_vector_type(8)));

#define NACC 11          // ce_num, vcount, psum1..3, gsum1..3, isum1..3
#define ROW_STRIDE 16    // padded partial row (floats)
#define P1_THREADS 256
#define P1_MAXBLOCKS 512

// Full wave32 allreduce-sum using one WMMA:
//   A[m][0] = x (lanes 0-15), A[m][2] = x (lanes 16-31), A[*][1]=A[*][3]=0 ; B = ones(4x16)
//   D[m][n] = x_m + x_{m+16}  (independent of n)
//   lane<16 holds D[0..7][n] (8 VGPRs), lane>=16 holds D[8..15][n]
//   sum 8 components, then add partner half via shfl_xor(16) -> every lane has the total.
__device__ __forceinline__ float wave_allsum(float x) {
  v2f a; a[0] = x;    a[1] = 0.0f;
  v2f b; b[0] = 1.0f; b[1] = 1.0f;
  v8f c = {};
  c = __builtin_amdgcn_wmma_f32_16x16x4_f32(false, a, false, b, (short)0, c, false, false);
  float s = ((c[0] + c[1]) + (c[2] + c[3])) + ((c[4] + c[5]) + (c[6] + c[7]));
  s += __shfl_xor(s, 16);
  return s;
}

// Branch-free per-voxel accumulation (4-class log-softmax + masked sums).
__device__ __forceinline__ void acc_voxel(float x0, float x1, float x2, float x3,
                                          int t, float acc[NACC]) {
  float m   = fmaxf(fmaxf(x0, x1), fmaxf(x2, x3));
  float e0  = __expf(x0 - m), e1 = __expf(x1 - m), e2 = __expf(x2 - m), e3 = __expf(x3 - m);
  float sum = (e0 + e1) + (e2 + e3);
  float lse = m + __logf(sum);
  float inv = 1.0f / sum;

  float vf = (t != -1) ? 1.0f : 0.0f;
  int   t0 = (t != -1) ? t : 0;

  float xt = x0;
  xt = (t0 == 1) ? x1 : xt;
  xt = (t0 == 2) ? x2 : xt;
  xt = (t0 == 3) ? x3 : xt;

  acc[0] += vf * (lse - xt);   // CE numerator (masked NLL)
  acc[1] += vf;                // valid count

  float p1 = e1 * inv, p2 = e2 * inv, p3 = e3 * inv;
  acc[2] += vf * p1;           // p_sum, classes 1..3
  acc[3] += vf * p2;
  acc[4] += vf * p3;

  float g1 = (t0 == 1) ? vf : 0.0f;
  float g2 = (t0 == 2) ? vf : 0.0f;
  float g3 = (t0 == 3) ? vf : 0.0f;
  acc[5] += g1;                // gt_sum, classes 1..3
  acc[6] += g2;
  acc[7] += g3;
  acc[8]  += g1 * p1;          // intersect, classes 1..3
  acc[9]  += g2 * p2;
  acc[10] += g3 * p3;
}

__global__ void __launch_bounds__(P1_THREADS)
dicece_partial_kernel(const float* __restrict__ logits,
                      const int*   __restrict__ target,
                      float* __restrict__ partials,
                      long S,        // per-batch spatial volume (64*256*256)
                      long N4) {     // total voxels / 4
  float acc[NACC];
#pragma unroll
  for (int j = 0; j < NACC; ++j) acc[j] = 0.0f;

  const long gid = (long)blockIdx.x * blockDim.x + threadIdx.x;
  const long gsz = (long)gridDim.x * blockDim.x;

  for (long i = gid; i < N4; i += gsz) {
    long v = i << 2;                         // voxel base index (multiple of 4)
    int  b = (v >= S) ? 1 : 0;               // batch (B == 2)
    long s = v - (long)b * S;                // spatial index within batch
    const float* base = logits + (long)(b * 4) * S + s;

    float4 x0 = *(const float4*)(base);
    float4 x1 = *(const float4*)(base + S);
    float4 x2 = *(const float4*)(base + 2 * S);
    float4 x3 = *(const float4*)(base + 3 * S);
    int4   tt = *(const int4*)(target + v);

    acc_voxel(x0.x, x1.x, x2.x, x3.x, tt.x, acc);
    acc_voxel(x0.y, x1.y, x2.y, x3.y, tt.y, acc);
    acc_voxel(x0.z, x1.z, x2.z, x3.z, tt.z, acc);
    acc_voxel(x0.w, x1.w, x2.w, x3.w, tt.w, acc);
  }

  // Wave32 allreduce via WMMA (EXEC all-ones here: loop fully reconverged).
  float wsum[NACC];
#pragma unroll
  for (int j = 0; j < NACC; ++j) wsum[j] = wave_allsum(acc[j]);

  __shared__ float sm[P1_THREADS / 32][NACC];
  const int lane = threadIdx.x & 31;
  const int wave = threadIdx.x >> 5;
  if (lane == 0) {
#pragma unroll
    for (int j = 0; j < NACC; ++j) sm[wave][j] = wsum[j];
  }
  __syncthreads();

  if (threadIdx.x < NACC) {
    float t = 0.0f;
#pragma unroll
    for (int w = 0; w < P1_THREADS / 32; ++w) t += sm[w][threadIdx.x];
    partials[(long)blockIdx.x * ROW_STRIDE + threadIdx.x] = t;
  }
}

__global__ void __launch_bounds__(P1_THREADS)
dicece_finalize_kernel(const float* __restrict__ partials,
                       int nrows,
                       float* __restrict__ out) {
  float acc[NACC];
#pragma unroll
  for (int j = 0; j < NACC; ++j) acc[j] = 0.0f;

  for (int r = threadIdx.x; r < nrows; r += blockDim.x) {
#pragma unroll
    for (int j = 0; j < NACC; ++j) acc[j] += partials[(long)r * ROW_STRIDE + j];
  }

  float wsum[NACC];
#pragma unroll
  for (int j = 0; j < NACC; ++j) wsum[j] = wave_allsum(acc[j]);

  __shared__ float sm[P1_THREADS / 32][NACC];
  const int lane = threadIdx.x & 31;
  const int wave = threadIdx.x >> 5;
  if (lane == 0) {
#pragma unroll
    for (int j = 0; j < NACC; ++j) sm[wave][j] = wsum[j];
  }
  __syncthreads();

  if (threadIdx.x == 0) {
    float a[NACC];
#pragma unroll
    for (int j = 0; j < NACC; ++j) {
      float t = 0.0f;
#pragma unroll
      for (int w = 0; w < P1_THREADS / 32; ++w) t += sm[w][j];
      a[j] = t;
    }
    const float SNR = 1e-5f, SDR = 1e-5f;
    float ce = a[0] / a[1];
    float d1 = (2.0f * a[8]  + SNR) / (a[2] + a[5] + SDR);
    float d2 = (2.0f * a[9]  + SNR) / (a[3] + a[6] + SDR);
    float d3 = (2.0f * a[10] + SNR) / (a[4] + a[7] + SDR);
    float pr1 = (a[5] > 0.0f) ? 1.0f : 0.0f;
    float pr2 = (a[6] > 0.0f) ? 1.0f : 0.0f;
    float pr3 = (a[7] > 0.0f) ? 1.0f : 0.0f;
    float np  = pr1 + pr2 + pr3;
    float md  = (d1 * pr1 + d2 * pr2 + d3 * pr3) / fmaxf(np, 1.0f);
    float dl  = (np > 0.0f) ? (1.0f - md) : 0.0f;
    out[0] = dl + ce;   // DICE_W = CE_W = 1
  }
}

extern "C" void kernel_launch(void* const* d_in, const int* in_sizes, int n_in,
                              void* d_out, int out_size, void* d_ws, size_t ws_size,
                              hipStream_t stream) {
  const float* logits = (const float*)d_in[0];
  const int*   target = (const int*)d_in[1];
  float*       out    = (float*)d_out;
  float*       parts  = (float*)d_ws;

  const long Nv = (long)in_sizes[1];   // total voxels = B * D*H*W = 8,388,608
  const long S  = Nv / 2;              // per-batch spatial volume (B == 2)
  const long N4 = Nv >> 2;

  int blocks = P1_MAXBLOCKS;
  long maxRows = (long)(ws_size / (ROW_STRIDE * sizeof(float)));
  if (maxRows < blocks) blocks = (int)maxRows;
  if (blocks < 1) blocks = 1;

  dicece_partial_kernel<<<blocks, P1_THREADS, 0, stream>>>(logits, target, parts, S, N4);
  dicece_finalize_kernel<<<1, P1_THREADS, 0, stream>>>(parts, blocks, out);
}